// VolumeTransform_3667902071054
// MI455X (gfx1250) — compile-verified
//
#include <hip/hip_runtime.h>
#include <hip/hip_bf16.h>

typedef __attribute__((ext_vector_type(2))) float v2f;
typedef __attribute__((ext_vector_type(8))) float v8f;

namespace {
constexpr int D_OCC = 16;
constexpr int H_OCC = 200;
constexpr int W_OCC = 200;
constexpr int PLANE = D_OCC * H_OCC * W_OCC;   // 640000 elems per channel
constexpr int FH = 112;                        // 900/8
constexpr int FW = 200;                        // 1600/8
constexpr int FD = 56;                         // int((58-2)//1)
constexpr int WBLK = 16;                       // pixels per wave
constexpr int NWB  = (FW + WBLK - 1) / WBLK;   // 13
constexpr int NPB  = FH * NWB;                 // 1456 pixel-blocks (one per wave)
constexpr int WAVES_PER_BLOCK = 8;             // 256 threads
}

__global__ __launch_bounds__(256) void volume_transform_kernel(
    const float* __restrict__ vol,   // (1,32,16,200,200)
    const float* __restrict__ Km,    // (1,3,3)
    const float* __restrict__ Tm,    // (1,4,4)
    float* __restrict__ out)         // (1,32,112,200)
{
  const int tid  = threadIdx.x;
  const int wave = tid >> 5;
  const int lane = tid & 31;
  const int p    = lane & 15;   // pixel within 16-wide strip
  const int half = lane >> 4;   // channel group: 0 -> ch 0..15, 1 -> ch 16..31

  const int pb = blockIdx.x * WAVES_PER_BLOCK + wave;
  if (pb >= NPB) return;                     // wave-uniform exit
  const int h  = pb / NWB;
  const int wb = pb % NWB;
  const int w  = wb * WBLK + p;              // output column (guard at store)

  // ---------------- fused projection matrix M (rows 0..2), wave-uniform ----
  // cam2img = [[K,0],[0,1]]  =>  img2cam = [[K^-1,0],[0,1]] (analytic adjugate)
  const float K00 = Km[0], K01 = Km[1], K02 = Km[2];
  const float K10 = Km[3], K11 = Km[4], K12 = Km[5];
  const float K20 = Km[6], K21 = Km[7], K22 = Km[8];
  const float c00 =  (K11 * K22 - K12 * K21);
  const float c01 = -(K10 * K22 - K12 * K20);
  const float c02 =  (K10 * K21 - K11 * K20);
  const float rdet = 1.0f / (K00 * c00 + K01 * c01 + K02 * c02);
  const float i00 = c00 * rdet;
  const float i01 = (K02 * K21 - K01 * K22) * rdet;
  const float i02 = (K01 * K12 - K02 * K11) * rdet;
  const float i10 = c01 * rdet;
  const float i11 = (K00 * K22 - K02 * K20) * rdet;
  const float i12 = (K02 * K10 - K00 * K12) * rdet;
  const float i20 = c02 * rdet;
  const float i21 = (K01 * K20 - K00 * K21) * rdet;
  const float i22 = (K00 * K11 - K01 * K10) * rdet;

  // e2f: scale 2.5 on x/y/z, translation (100, 100, 2.5)  [ratios are all 1]
  // P = T @ img2cam ;  M[i][j] = 2.5*P[i][j] + t_i * P[3][j]
  const float T30 = Tm[12], T31 = Tm[13], T32 = Tm[14], T33 = Tm[15];
  const float P30 = T30 * i00 + T31 * i10 + T32 * i20;
  const float P31 = T30 * i01 + T31 * i11 + T32 * i21;
  const float P32 = T30 * i02 + T31 * i12 + T32 * i22;
  const float P33 = T33;

  float M00, M01, M02, M03, M10, M11, M12, M13, M20, M21, M22, M23;
  {
    const float tv0 = 100.0f, tv1 = 100.0f, tv2 = 2.5f;
    float Ti0, Ti1, Ti2, Ti3;
    Ti0 = Tm[0]; Ti1 = Tm[1]; Ti2 = Tm[2]; Ti3 = Tm[3];
    M00 = 2.5f * (Ti0 * i00 + Ti1 * i10 + Ti2 * i20) + tv0 * P30;
    M01 = 2.5f * (Ti0 * i01 + Ti1 * i11 + Ti2 * i21) + tv0 * P31;
    M02 = 2.5f * (Ti0 * i02 + Ti1 * i12 + Ti2 * i22) + tv0 * P32;
    M03 = 2.5f * Ti3 + tv0 * P33;
    Ti0 = Tm[4]; Ti1 = Tm[5]; Ti2 = Tm[6]; Ti3 = Tm[7];
    M10 = 2.5f * (Ti0 * i00 + Ti1 * i10 + Ti2 * i20) + tv1 * P30;
    M11 = 2.5f * (Ti0 * i01 + Ti1 * i11 + Ti2 * i21) + tv1 * P31;
    M12 = 2.5f * (Ti0 * i02 + Ti1 * i12 + Ti2 * i22) + tv1 * P32;
    M13 = 2.5f * Ti3 + tv1 * P33;
    Ti0 = Tm[8]; Ti1 = Tm[9]; Ti2 = Tm[10]; Ti3 = Tm[11];
    M20 = 2.5f * (Ti0 * i00 + Ti1 * i10 + Ti2 * i20) + tv2 * P30;
    M21 = 2.5f * (Ti0 * i01 + Ti1 * i11 + Ti2 * i21) + tv2 * P31;
    M22 = 2.5f * (Ti0 * i02 + Ti1 * i12 + Ti2 * i22) + tv2 * P32;
    M23 = 2.5f * Ti3 + tv2 * P33;
  }

  // ---------------- ray setup via V_WMMA_F32_16X16X4_F32 --------------------
  // A (16x4): rows {0,1,2} and {8,9,10} = M[:, :3] (col3 = 0), other rows 0.
  //   Duplicating rows means D VGPR r carries row r (lanes 0-15) AND row r+8
  //   (lanes 16-31) -> ax/ay/az land in every lane with no cross-lane fixup.
  // B (4x16): column n = [px_n, py_n, 1, 0]. Layout mirrors documented A
  //   layout: VGPR0 = K0 (lanes 0-15) / K2 (lanes 16-31), VGPR1 = K1 / K3.
  const float XSTEP = 1599.0f / 199.0f;   // linspace(0,1599,200)
  const float YSTEP = 899.0f / 111.0f;    // linspace(0, 899,112)
  const float px = (float)w * XSTEP;
  const float py = (float)h * YSTEP;

  const int  e      = p & 7;
  const bool hasRow = (e < 3);
  float a0, a1;
  if (half == 0) {
    a0 = hasRow ? ((e == 0) ? M00 : (e == 1) ? M10 : M20) : 0.0f;  // K=0 col
    a1 = hasRow ? ((e == 0) ? M01 : (e == 1) ? M11 : M21) : 0.0f;  // K=1 col
  } else {
    a0 = hasRow ? ((e == 0) ? M02 : (e == 1) ? M12 : M22) : 0.0f;  // K=2 col
    a1 = 0.0f;                                                     // K=3 col
  }
  v2f Aop; Aop.x = a0; Aop.y = a1;
  v2f Bop;
  Bop.x = (half == 0) ? px : 1.0f;   // K=0 : K=2
  Bop.y = (half == 0) ? py : 0.0f;   // K=1 : K=3

  v8f Cz = {};
  v8f Dm = __builtin_amdgcn_wmma_f32_16x16x4_f32(
      false, Aop, false, Bop, (short)0, Cz, false, false);
  const float ax = Dm[0], ay = Dm[1], az = Dm[2];
  const float tx = M03,  ty = M13,  tz = M23;

  // ---------------- depth march + trilinear gather --------------------------
  float acc[16];
#pragma unroll
  for (int j = 0; j < 16; ++j) acc[j] = 0.0f;

  const float* __restrict__ volc = vol + half * 16 * PLANE;
  const float DSTEP = 56.0f / 55.0f;       // linspace(2,58,56)

  for (int k = 0; k < FD; ++k) {
    const float dv = fmaf((float)k, DSTEP, 2.0f);
    // normalization cancels: voxel coords == M @ pts
    const float vx = fmaf(ax, dv, tx);     // -> W
    const float vy = fmaf(ay, dv, ty);     // -> H
    const float vz = fmaf(az, dv, tz);     // -> D
    const float fx = floorf(vx), fy = floorf(vy), fz = floorf(vz);
    const float rx = vx - fx, ry = vy - fy, rz = vz - fz;
    const int ix = (int)fx, iy = (int)fy, iz = (int)fz;

    // zero-padding: fold per-axis validity into the 1D weights
    const float mx0 = (ix   >= 0 && ix   < W_OCC) ? 1.0f : 0.0f;
    const float mx1 = (ix+1 >= 0 && ix+1 < W_OCC) ? 1.0f : 0.0f;
    const float my0 = (iy   >= 0 && iy   < H_OCC) ? 1.0f : 0.0f;
    const float my1 = (iy+1 >= 0 && iy+1 < H_OCC) ? 1.0f : 0.0f;
    const float mz0 = (iz   >= 0 && iz   < D_OCC) ? 1.0f : 0.0f;
    const float mz1 = (iz+1 >= 0 && iz+1 < D_OCC) ? 1.0f : 0.0f;

    const bool inside = ((mx0 + mx1) > 0.0f) & ((my0 + my1) > 0.0f) &
                        ((mz0 + mz1) > 0.0f);
    if (!__any(inside)) continue;          // wave-uniform skip, EXEC untouched

    const float wx0 = (1.0f - rx) * mx0, wx1 = rx * mx1;
    const float wy0 = (1.0f - ry) * my0, wy1 = ry * my1;
    const float wz0 = (1.0f - rz) * mz0, wz1 = rz * mz1;
    const float w000 = wz0*wy0*wx0, w001 = wz0*wy0*wx1;
    const float w010 = wz0*wy1*wx0, w011 = wz0*wy1*wx1;
    const float w100 = wz1*wy0*wx0, w101 = wz1*wy0*wx1;
    const float w110 = wz1*wy1*wx0, w111 = wz1*wy1*wx1;

    const int cx0 = min(max(ix,   0), W_OCC-1), cx1 = min(max(ix+1, 0), W_OCC-1);
    const int cy0 = min(max(iy,   0), H_OCC-1), cy1 = min(max(iy+1, 0), H_OCC-1);
    const int cz0 = min(max(iz,   0), D_OCC-1), cz1 = min(max(iz+1, 0), D_OCC-1);

    const int r00 = (cz0 * H_OCC + cy0) * W_OCC;
    const int r01 = (cz0 * H_OCC + cy1) * W_OCC;
    const int r10 = (cz1 * H_OCC + cy0) * W_OCC;
    const int r11 = (cz1 * H_OCC + cy1) * W_OCC;
    const int o000 = r00 + cx0, o001 = r00 + cx1;
    const int o010 = r01 + cx0, o011 = r01 + cx1;
    const int o100 = r10 + cx0, o101 = r10 + cx1;
    const int o110 = r11 + cx0, o111 = r11 + cx1;

#pragma unroll
    for (int j = 0; j < 16; ++j) {
      const float* __restrict__ pj = volc + j * PLANE;  // L2-resident gathers
      const float g000 = pj[o000], g001 = pj[o001];
      const float g010 = pj[o010], g011 = pj[o011];
      const float g100 = pj[o100], g101 = pj[o101];
      const float g110 = pj[o110], g111 = pj[o111];
      float s = g000 * w000;
      s = fmaf(g001, w001, s);
      s = fmaf(g010, w010, s);
      s = fmaf(g011, w011, s);
      s = fmaf(g100, w100, s);
      s = fmaf(g101, w101, s);
      s = fmaf(g110, w110, s);
      s = fmaf(g111, w111, s);
      acc[j] += s;
    }
  }

  // ---------------- store: out[c, h, w], c = half*16 + j --------------------
  if (w < FW) {
    float* __restrict__ op = out + (half * 16) * (FH * FW) + h * FW + w;
#pragma unroll
    for (int j = 0; j < 16; ++j) {
      op[j * (FH * FW)] = acc[j];
    }
  }
}

extern "C" void kernel_launch(void* const* d_in, const int* in_sizes, int n_in,
                              void* d_out, int out_size, void* d_ws, size_t ws_size,
                              hipStream_t stream) {
  const float* vol = (const float*)d_in[0];  // occ_feature (1,32,16,200,200) f32
  const float* Km  = (const float*)d_in[1];  // K (1,3,3) f32
  const float* Tm  = (const float*)d_in[2];  // T (1,4,4) f32
  float* out = (float*)d_out;                // (1,32,112,200) f32

  const int blocks = (NPB + WAVES_PER_BLOCK - 1) / WAVES_PER_BLOCK;  // 182
  volume_transform_kernel<<<blocks, 256, 0, stream>>>(vol, Km, Tm, out);
}